// Encoder_GCN_30013231464613
// MI455X (gfx1250) — compile-verified
//
#include <hip/hip_runtime.h>

#define N_NODES  50000
#define N_EDGES  800000
#define DIM      128
#define N_GRAPHS 128
#define N_LAYERS 3
#define XS_DIM   (DIM * N_LAYERS)          // 384
#define POOL_ELEMS (N_GRAPHS * XS_DIM)     // 49152
#define LDS_PITCH 132                      // pad 128 -> 132 for bank-conflict-free WMMA reads

typedef __attribute__((ext_vector_type(2))) float v2f;
typedef __attribute__((ext_vector_type(8))) float v8f;

// ---------------------------------------------------------------- utilities
__global__ void fill_f32(float* __restrict__ p, float v, int n) {
  int t = blockIdx.x * blockDim.x + threadIdx.x;
  if (t < n) p[t] = v;
}

__global__ void deg_accum(const int* __restrict__ dst, float* __restrict__ deg) {
  int e = blockIdx.x * blockDim.x + threadIdx.x;
  if (e < N_EDGES) atomicAdd(&deg[dst[e]], 1.0f);
}

__global__ void rsqrt_inplace(float* __restrict__ p, int n) {
  int t = blockIdx.x * blockDim.x + threadIdx.x;
  if (t < n) p[t] = rsqrtf(p[t]);   // deg >= 1 always (self-loops)
}

__global__ void edge_norm(const int* __restrict__ src, const int* __restrict__ dst,
                          const float* __restrict__ dinv, float* __restrict__ nrm) {
  int e = blockIdx.x * blockDim.x + threadIdx.x;
  if (e < N_EDGES) nrm[e] = dinv[src[e]] * dinv[dst[e]];
}

__global__ void count_nodes(const int* __restrict__ batch, float* __restrict__ counts) {
  int i = blockIdx.x * blockDim.x + threadIdx.x;
  if (i < N_NODES) atomicAdd(&counts[batch[i]], 1.0f);
}

// ------------------------------------------------ dense GEMM via f32 WMMA
// C[16x128 tile] = A[16x128 rows] @ W[128x128].  One block = one 16-row tile,
// 8 waves, wave w owns the 16-column slab n0 = 16*w.  K-loop: 32 steps of
// V_WMMA_F32_16X16X4_F32 (K=4 each) over the 128-deep reduction.
// A tile staged in LDS (coalesced load, padded pitch for conflict-free reads).
__global__ void __launch_bounds__(256)
gcn_gemm_wmma(const float* __restrict__ A, int lda,
              const float* __restrict__ W, float* __restrict__ out) {
  __shared__ float As[16 * LDS_PITCH];
  const int rowBase = blockIdx.x * 16;   // N_NODES = 3125 * 16 exactly

  // cooperative, coalesced A-tile load: 2048 floats / 256 threads = 8 each
  #pragma unroll
  for (int i = 0; i < 8; ++i) {
    int idx = threadIdx.x + i * 256;
    int r = idx >> 7, c = idx & 127;
    As[r * LDS_PITCH + c] = A[(rowBase + r) * lda + c];
  }
  __syncthreads();

  const int wave = threadIdx.x >> 5;
  const int lane = threadIdx.x & 31;
  const int m    = lane & 15;        // matrix row (A) / column index (B,D)
  const int half = lane >> 4;        // 0: K even base, 1: K odd pair (+2)
  const int n    = wave * 16 + m;    // output column owned by this lane

  v8f acc = {};
  for (int kk = 0; kk < 32; ++kk) {
    const int k = 4 * kk + 2 * half;
    // A 16x4 fragment: VGPR0 = K {0,2}, VGPR1 = K {1,3}
    v2f a;
    a.x = As[m * LDS_PITCH + k];
    a.y = As[m * LDS_PITCH + k + 1];
    // B 4x16 fragment: rows of W striped across lanes (same K split as A)
    v2f b;
    b.x = W[(k    ) * DIM + n];
    b.y = W[(k + 1) * DIM + n];
    acc = __builtin_amdgcn_wmma_f32_16x16x4_f32(
        /*neg_a=*/false, a, /*neg_b=*/false, b,
        /*c_mod=*/(short)0, acc, /*reuse_a=*/false, /*reuse_b=*/false);
  }

  // D layout: VGPR v holds row (v + 8*half), column n
  #pragma unroll
  for (int v = 0; v < 8; ++v)
    out[(rowBase + v + 8 * half) * DIM + n] = acc[v];
}

// --------------------------------------- aggregation: self-loop + edge scatter
__global__ void selfloop_init(const float* __restrict__ hlin,
                              const float* __restrict__ dinv,
                              float* __restrict__ agg) {
  int tid = blockIdx.x * blockDim.x + threadIdx.x;
  int i = tid >> 5;
  if (i >= N_NODES) return;
  int c = (tid & 31) * 4;
  float d = dinv[i];
  float w = d * d;                       // self-loop norm = dinv[i]^2
  float4 v = *(const float4*)(hlin + i * DIM + c);
  v.x *= w; v.y *= w; v.z *= w; v.w *= w;
  *(float4*)(agg + i * DIM + c) = v;
}

// one wave per edge: float4 gather of h[src], 4x f32 atomic scatter to agg[dst]
__global__ void gcn_edge_scatter(const int* __restrict__ src,
                                 const int* __restrict__ dst,
                                 const float* __restrict__ nrm,
                                 const float* __restrict__ hlin,
                                 float* __restrict__ agg) {
  int tid = blockIdx.x * blockDim.x + threadIdx.x;
  int e = tid >> 5;
  if (e >= N_EDGES) return;
  int lane = tid & 31;
  int s = src[e], d = dst[e];            // wave-uniform broadcast loads
  float w = nrm[e];
  float4 v = *(const float4*)(hlin + s * DIM + lane * 4);
  float* ap = agg + d * DIM + lane * 4;
  atomicAdd(ap + 0, v.x * w);
  atomicAdd(ap + 1, v.y * w);
  atomicAdd(ap + 2, v.z * w);
  atomicAdd(ap + 3, v.w * w);
}

// bias + ReLU, write xs slab of d_out, accumulate per-graph pool sums
__global__ void gcn_bias_relu_pool(const float* __restrict__ agg,
                                   const float* __restrict__ bias,
                                   const int* __restrict__ batch,
                                   float* __restrict__ out, int layer) {
  int tid = blockIdx.x * blockDim.x + threadIdx.x;
  int i = tid >> 5;
  if (i >= N_NODES) return;
  int c = (tid & 31) * 4;
  float4 v = *(const float4*)(agg + i * DIM + c);
  float4 b = *(const float4*)(bias + c);
  v.x = fmaxf(v.x + b.x, 0.0f);
  v.y = fmaxf(v.y + b.y, 0.0f);
  v.z = fmaxf(v.z + b.z, 0.0f);
  v.w = fmaxf(v.w + b.w, 0.0f);
  // xs output (second part of d_out), row-major [N_NODES, 384]
  *(float4*)(out + POOL_ELEMS + i * XS_DIM + layer * DIM + c) = v;
  // pool accumulation into d_out[:POOL_ELEMS]
  int g = batch[i];
  float* pr = out + g * XS_DIM + layer * DIM + c;
  atomicAdd(pr + 0, v.x);
  atomicAdd(pr + 1, v.y);
  atomicAdd(pr + 2, v.z);
  atomicAdd(pr + 3, v.w);
}

__global__ void pool_finalize(float* __restrict__ pool, const float* __restrict__ counts) {
  int t = blockIdx.x * blockDim.x + threadIdx.x;
  if (t >= POOL_ELEMS) return;
  float c = counts[t / XS_DIM];
  pool[t] /= fmaxf(c, 1.0f);
}

// ---------------------------------------------------------------- launcher
extern "C" void kernel_launch(void* const* d_in, const int* in_sizes, int n_in,
                              void* d_out, int out_size, void* d_ws, size_t ws_size,
                              hipStream_t stream) {
  const float* x     = (const float*)d_in[0];
  const int*   ei    = (const int*)d_in[1];          // [2, E]
  const int*   batch = (const int*)d_in[2];
  const float* W[3]  = {(const float*)d_in[3], (const float*)d_in[5], (const float*)d_in[7]};
  const float* B[3]  = {(const float*)d_in[4], (const float*)d_in[6], (const float*)d_in[8]};
  float* out = (float*)d_out;
  float* ws  = (float*)d_ws;

  const int* src = ei;
  const int* dst = ei + N_EDGES;

  float* deg    = ws;                                 // 50000 (becomes dinv)
  float* nrm    = ws + N_NODES;                       // 800000
  float* hlin   = ws + N_NODES + N_EDGES;             // 6.4M
  float* agg    = hlin + (size_t)N_NODES * DIM;       // 6.4M
  float* counts = agg  + (size_t)N_NODES * DIM;       // 128

  const int T = 256;
  // --- normalization precompute -------------------------------------------
  fill_f32<<<(N_NODES + T - 1) / T, T, 0, stream>>>(deg, 1.0f, N_NODES);
  deg_accum<<<(N_EDGES + T - 1) / T, T, 0, stream>>>(dst, deg);
  rsqrt_inplace<<<(N_NODES + T - 1) / T, T, 0, stream>>>(deg, N_NODES);
  edge_norm<<<(N_EDGES + T - 1) / T, T, 0, stream>>>(src, dst, deg, nrm);
  // --- pooling precompute --------------------------------------------------
  fill_f32<<<1, N_GRAPHS, 0, stream>>>(counts, 0.0f, N_GRAPHS);
  count_nodes<<<(N_NODES + T - 1) / T, T, 0, stream>>>(batch, counts);
  fill_f32<<<(POOL_ELEMS + T - 1) / T, T, 0, stream>>>(out, 0.0f, POOL_ELEMS);

  // --- 3 GCN layers --------------------------------------------------------
  for (int l = 0; l < N_LAYERS; ++l) {
    const float* Ain = (l == 0) ? x : (out + POOL_ELEMS + (l - 1) * DIM);
    const int lda    = (l == 0) ? DIM : XS_DIM;
    gcn_gemm_wmma<<<N_NODES / 16, 256, 0, stream>>>(Ain, lda, W[l], hlin);
    selfloop_init<<<(N_NODES * 32 + T - 1) / T, T, 0, stream>>>(hlin, deg, agg);
    gcn_edge_scatter<<<(N_EDGES * 32) / T, T, 0, stream>>>(src, dst, nrm, hlin, agg);
    gcn_bias_relu_pool<<<(N_NODES * 32 + T - 1) / T, T, 0, stream>>>(agg, B[l], batch, out, l);
  }
  pool_finalize<<<(POOL_ELEMS + T - 1) / T, T, 0, stream>>>(out, counts);
}